// RegionDiscriminativeLoss_32229434589498
// MI455X (gfx1250) — compile-verified
//
#include <hip/hip_runtime.h>
#include <hip/hip_bf16.h>
#include <math.h>

// Problem constants (fixed by setup_inputs)
#define NIMG   4
#define CCH    128
#define HW     (256*512)          // 131072 pixels per image
#define KCLS   19
#define THEA   0.5f
#define DELTA  1.5f
#define MIN_PIX 20.0f
#define MSTR   129                // padded LDS row stride (bank spread)

typedef __attribute__((ext_vector_type(2))) float v2f;
typedef __attribute__((ext_vector_type(8))) float v8f;

// ---------------- K0: zero accumulator region of workspace ----------------
__global__ void k0_zero(float* __restrict__ ws, int nfloats) {
    int i = blockIdx.x * blockDim.x + threadIdx.x;
    if (i < nfloats) ws[i] = 0.0f;
}

// ---------------- K2: segment sums + counts (streaming pass 1) -------------
// grid = (HW/2048, NIMG), block = 256.  Each block: 2048-pixel tile, all 128
// channel planes (contiguous loads), LDS atomic accumulation, global flush.
__global__ void k2_segsum(const float* __restrict__ x, const int* __restrict__ lab,
                          float* __restrict__ sums, float* __restrict__ counts) {
    const int tile = blockIdx.x;      // 0..63
    const int n    = blockIdx.y;      // 0..3
    const int t    = threadIdx.x;     // 0..255

    __shared__ float slds[KCLS * MSTR];   // 9804 B
    __shared__ float clds[KCLS];
    for (int i = t; i < KCLS * MSTR; i += 256) slds[i] = 0.0f;
    if (t < KCLS) clds[t] = 0.0f;
    __syncthreads();

    const int p0 = tile * 2048;
    // labels: thread t owns pixels p0 + 4t..4t+3 and p0 + 1024 + 4t..4t+3
    const int4* lb4 = (const int4*)(lab + (size_t)n * HW + p0);
    int4 la = lb4[t];
    int4 lc = lb4[256 + t];
    int kr[8] = { la.x, la.y, la.z, la.w, lc.x, lc.y, lc.z, lc.w };
    #pragma unroll
    for (int s = 0; s < 8; ++s) {
        if ((unsigned)kr[s] >= (unsigned)KCLS) kr[s] = -1;   // IGNORE / OOR
        else atomicAdd(&clds[kr[s]], 1.0f);
    }

    const float* pbase = x + ((size_t)n * CCH) * HW + p0;
    for (int c = 0; c < CCH; ++c) {
        const float4* p4 = (const float4*)(pbase + (size_t)c * HW);
        if (c + 2 < CCH)
            __builtin_prefetch(pbase + (size_t)(c + 2) * HW + t * 4, 0, 1);
        float4 va = p4[t];
        float4 vb = p4[256 + t];
        float xv[8] = { va.x, va.y, va.z, va.w, vb.x, vb.y, vb.z, vb.w };
        #pragma unroll
        for (int s = 0; s < 8; ++s)
            if (kr[s] >= 0) atomicAdd(&slds[kr[s] * MSTR + c], xv[s]);
    }
    __syncthreads();

    for (int i = t; i < KCLS * CCH; i += 256) {
        int k = i >> 7, c = i & 127;
        float v = slds[k * MSTR + c];
        if (v != 0.0f) atomicAdd(&sums[((size_t)n * KCLS + k) * CCH + c], v);
    }
    if (t < KCLS && clds[t] != 0.0f) atomicAdd(&counts[n * KCLS + t], clds[t]);
}

// ---------------- K3: centers + Gram (WMMA) + loss_dis/loss_reg ------------
// grid = NIMG, block = 128 (wave 0 runs the WMMA Gram).
// Gram G = M * M^T with M = [32 x 128] f32 (rows >= KCLS zero), via
// V_WMMA_F32_16X16X4_F32: 2x2 output tiles, K-loop over 128 in steps of 4.
__global__ void k3_centers(const float* __restrict__ sums, const float* __restrict__ counts,
                           float* __restrict__ means_g, float* __restrict__ part_g) {
    const int n = blockIdx.x;
    const int t = threadIdx.x;     // 0..127
    __shared__ float mlds[32 * MSTR];     // padded 32x128 center matrix
    __shared__ float G[32 * 33];          // Gram (padded stride 33)

    for (int i = t; i < 32 * MSTR; i += 128) mlds[i] = 0.0f;
    __syncthreads();

    // means: thread t = channel, loop classes
    for (int k = 0; k < KCLS; ++k) {
        float cnt = counts[n * KCLS + k];
        float inv = 1.0f / fmaxf(cnt, 1.0f);
        float m = sums[((size_t)n * KCLS + k) * CCH + t] * inv;
        mlds[k * MSTR + t] = m;
        means_g[((size_t)n * KCLS + k) * CCH + t] = m;
    }
    __syncthreads();

    if (t < 32) {                          // wave 0, EXEC all ones
        const int lmod  = t & 15;          // M row within tile (A) / N col (B,D)
        const int lhalf = t >> 4;          // selects K pair {0,1} vs {2,3}
        #pragma unroll
        for (int mt = 0; mt < 2; ++mt) {
            #pragma unroll
            for (int nt = 0; nt < 2; ++nt) {
                v8f acc = {};
                for (int kk = 0; kk < CCH; kk += 4) {
                    const int ka = kk + 2 * lhalf;
                    v2f a, b;
                    a.x = mlds[(mt * 16 + lmod) * MSTR + ka];
                    a.y = mlds[(mt * 16 + lmod) * MSTR + ka + 1];
                    b.x = mlds[(nt * 16 + lmod) * MSTR + ka];   // B = A^T source
                    b.y = mlds[(nt * 16 + lmod) * MSTR + ka + 1];
                    acc = __builtin_amdgcn_wmma_f32_16x16x4_f32(
                        false, a, false, b, (short)0, acc, false, false);
                }
                #pragma unroll
                for (int j = 0; j < 8; ++j) {
                    int gm = mt * 16 + j + 8 * lhalf;   // D row layout
                    int gn = nt * 16 + lmod;            // D col layout
                    G[gm * 33 + gn] = acc[j];
                }
            }
        }
    }
    __syncthreads();

    if (t == 0) {                          // tiny 19x19 epilogue, serial
        float vflag[KCLS], mn2[KCLS];
        float nv = 0.0f;
        for (int k = 0; k < KCLS; ++k) {
            float c0 = counts[n * KCLS + k];
            vflag[k] = (c0 > MIN_PIX) ? 1.0f : 0.0f;
            nv += vflag[k];
            mn2[k] = G[k * 33 + k];
        }
        float nvc = fmaxf(nv, 1.0f);
        float dis = 0.0f, reg = 0.0f;
        for (int f = 0; f < KCLS; ++f) {
            if (vflag[f] == 0.0f) continue;
            reg += sqrtf(fmaxf(mn2[f], 0.0f));
            for (int s = 0; s < KCLS; ++s) {
                if (s == f || vflag[s] == 0.0f) continue;
                float d2 = fmaxf(mn2[f] + mn2[s] - 2.0f * G[f * 33 + s], 0.0f);
                float D  = sqrtf(d2);
                float tr = fmaxf(2.0f * DELTA - D, 0.0f);
                dis += tr * tr;
            }
        }
        float connect = fmaxf(nvc * (nvc - 1.0f), 1.0f);
        part_g[n] = dis / connect + 0.001f * (reg / nvc);
    }
}

// ---------------- K4: loss_var (streaming pass 2) --------------------------
// grid = (HW/1024, NIMG), block = 256. Direct sum of (x - m)^2 per pixel.
__global__ void k4_var(const float* __restrict__ x, const int* __restrict__ lab,
                       const float* __restrict__ means_g, float* __restrict__ rsum) {
    const int tile = blockIdx.x;   // 0..127
    const int n    = blockIdx.y;
    const int t    = threadIdx.x;  // 0..255

    __shared__ float mlds[KCLS * MSTR];
    __shared__ float rlds[KCLS];
    for (int i = t; i < KCLS * CCH; i += 256) {
        int k = i >> 7, c = i & 127;
        mlds[k * MSTR + c] = means_g[((size_t)n * KCLS + k) * CCH + c];
    }
    if (t < KCLS) rlds[t] = 0.0f;
    __syncthreads();

    const int p0 = tile * 1024;
    const int4* lb4 = (const int4*)(lab + (size_t)n * HW + p0);
    int4 la = lb4[t];
    int kr[4] = { la.x, la.y, la.z, la.w };
    #pragma unroll
    for (int s = 0; s < 4; ++s)
        if ((unsigned)kr[s] >= (unsigned)KCLS) kr[s] = -1;

    float d2a[4] = {0.f, 0.f, 0.f, 0.f};
    const float* pbase = x + ((size_t)n * CCH) * HW + p0;
    for (int c = 0; c < CCH; ++c) {
        const float4* p4 = (const float4*)(pbase + (size_t)c * HW);
        if (c + 2 < CCH)
            __builtin_prefetch(pbase + (size_t)(c + 2) * HW + t * 4, 0, 1);
        float4 v = p4[t];
        float xv[4] = { v.x, v.y, v.z, v.w };
        #pragma unroll
        for (int s = 0; s < 4; ++s) {
            float m = (kr[s] >= 0) ? mlds[kr[s] * MSTR + c] : 0.0f;
            float d = xv[s] - m;
            d2a[s] = fmaf(d, d, d2a[s]);
        }
    }
    #pragma unroll
    for (int s = 0; s < 4; ++s) {
        if (kr[s] >= 0) {
            float dist = sqrtf(d2a[s]);
            float dr = fmaxf(dist - THEA, 0.0f);
            atomicAdd(&rlds[kr[s]], dr * dr);
        }
    }
    __syncthreads();
    if (t < KCLS && rlds[t] != 0.0f) atomicAdd(&rsum[n * KCLS + t], rlds[t]);
}

// ---------------- K5: combine -> scalar ------------------------------------
__global__ void k5_final(const float* __restrict__ counts, const float* __restrict__ rsum,
                         const float* __restrict__ part, float* __restrict__ out) {
    if (threadIdx.x == 0 && blockIdx.x == 0) {
        float total = 0.0f;
        for (int n = 0; n < NIMG; ++n) {
            float vflag[KCLS];
            float nv = 0.0f;
            for (int k = 0; k < KCLS; ++k) {
                float c0 = counts[n * KCLS + k];
                vflag[k] = (c0 > MIN_PIX) ? 1.0f : 0.0f;
                nv += vflag[k];
            }
            float nvc = fmaxf(nv, 1.0f);
            float lv = 0.0f;
            for (int k = 0; k < KCLS; ++k) {
                float c0 = counts[n * KCLS + k];
                lv += vflag[k] * (rsum[n * KCLS + k] / fmaxf(c0, 1.0f));
            }
            total += lv / nvc + part[n];
        }
        out[0] = total * 0.25f;
    }
}

// ---------------- launcher -------------------------------------------------
extern "C" void kernel_launch(void* const* d_in, const int* in_sizes, int n_in,
                              void* d_out, int out_size, void* d_ws, size_t ws_size,
                              hipStream_t stream) {
    const float* x  = (const float*)d_in[0];   // predict [4,128,256,512] f32
    const int* lab  = (const int*)d_in[1];     // target  [4,256,512] int
    float* out = (float*)d_out;
    float* ws  = (float*)d_ws;

    // workspace layout (floats)
    float* sums   = ws;            // 4*19*128 = 9728
    float* counts = ws + 9728;     // 76
    float* rsum   = ws + 9804;     // 76
    float* part   = ws + 9880;     // 4
    float* means  = ws + 9884;     // 9728 (written by K3, no zeroing needed)

    const int nzero = 9880;        // sums + counts + rsum (atomic accumulators)
    k0_zero<<<(nzero + 255) / 256, 256, 0, stream>>>(ws, nzero);
    k2_segsum<<<dim3(HW / 2048, NIMG), 256, 0, stream>>>(x, lab, sums, counts);
    k3_centers<<<NIMG, 128, 0, stream>>>(sums, counts, means, part);
    k4_var<<<dim3(HW / 1024, NIMG), 256, 0, stream>>>(x, lab, means, rsum);
    k5_final<<<1, 32, 0, stream>>>(counts, rsum, part, out);
    (void)in_sizes; (void)n_in; (void)out_size; (void)ws_size;
}